// KNNUpsample_29472065585610
// MI455X (gfx1250) — compile-verified
//
#include <hip/hip_runtime.h>

typedef __attribute__((ext_vector_type(2))) float v2f;
typedef __attribute__((ext_vector_type(4))) float v4f;
typedef __attribute__((ext_vector_type(8))) float v8f;

// Problem constants (from reference)
#define LD    16384
#define LU    65536
#define NN    8
#define D_IN  256
#define D_OUT 128
#define KNN   3
#define MROWS (LD * NN)        // 131072 MLP rows
#define UROWS (LU * NN)        // 524288 output rows
#define HS    132              // padded LDS row stride (floats), conflict-free

// ---------------------------------------------------------------------------
// Kernel 1: fused 2-layer MLP using V_WMMA_F32_16X16X4_F32.
// Grid: MROWS/16 blocks. Block: 256 threads = 8 waves; wave w owns output
// columns [16w, 16w+16). Layer-1 tile -> ReLU -> LDS -> Layer-2 tile -> H.
// ---------------------------------------------------------------------------
__global__ __launch_bounds__(256) void mlp2_kernel(
    const float* __restrict__ X,   // (MROWS, 256)
    const float* __restrict__ W1,  // (256, 128)
    const float* __restrict__ b1,  // (128)
    const float* __restrict__ W2,  // (128, 128)
    const float* __restrict__ b2,  // (128)
    float* __restrict__ H)         // (MROWS, 128)
{
    __shared__ float h1[16 * HS];  // 16 x 128 padded, 8448 B

    const int lane = threadIdx.x & 31;
    const int wave = threadIdx.x >> 5;     // 0..7 -> column tile
    const int half = lane >> 4;            // 0: K+{0,1}, 1: K+{2,3}
    const int lrow = lane & 15;            // M (A) or N (B/D) index
    const int n0   = wave * 16;            // tile column base
    const int row_base = blockIdx.x * 16;  // tile row base

    // ---- Layer 1: acc = X(16x256) * W1(256x16tile) + b1 ----
    v8f acc;
    {
        const float bias = b1[n0 + lrow];  // C/D: N depends only on lane%16
        #pragma unroll
        for (int r = 0; r < 8; ++r) acc[r] = bias;
    }
    const float* __restrict__ arow = X + (size_t)(row_base + lrow) * D_IN;
    #pragma unroll 4
    for (int k0 = 0; k0 < D_IN; k0 += 4) {
        const int kk = k0 + 2 * half;
        v2f a = *(const v2f*)(arow + kk);                  // A: M=lrow, K=kk,kk+1
        v2f b;
        b.x = W1[(size_t)kk * D_OUT + n0 + lrow];          // B: K=kk,   N=lrow
        b.y = W1[(size_t)(kk + 1) * D_OUT + n0 + lrow];    // B: K=kk+1, N=lrow
        acc = __builtin_amdgcn_wmma_f32_16x16x4_f32(
                  false, a, false, b, (short)0, acc, false, false);
    }

    // ReLU + stage to LDS (element (M = r + 8*half, N = lrow) per VGPR r)
    #pragma unroll
    for (int r = 0; r < 8; ++r)
        h1[(r + 8 * half) * HS + n0 + lrow] = fmaxf(acc[r], 0.0f);
    __syncthreads();

    // ---- Layer 2: acc2 = h1(16x128) * W2(128x16tile) + b2 ----
    v8f acc2;
    {
        const float bias = b2[n0 + lrow];
        #pragma unroll
        for (int r = 0; r < 8; ++r) acc2[r] = bias;
    }
    #pragma unroll 4
    for (int k0 = 0; k0 < D_OUT; k0 += 4) {
        const int kk = k0 + 2 * half;
        v2f a = *(const v2f*)(&h1[lrow * HS + kk]);        // ds_load_b64, conflict-free
        v2f b;
        b.x = W2[(size_t)kk * D_OUT + n0 + lrow];
        b.y = W2[(size_t)(kk + 1) * D_OUT + n0 + lrow];
        acc2 = __builtin_amdgcn_wmma_f32_16x16x4_f32(
                   false, a, false, b, (short)0, acc2, false, false);
    }

    // Store D tile: row = row_base + r + 8*half, col = n0 + lrow
    #pragma unroll
    for (int r = 0; r < 8; ++r)
        H[(size_t)(row_base + r + 8 * half) * D_OUT + n0 + lrow] = acc2[r];
}

// ---------------------------------------------------------------------------
// Kernel 2: out = up + (H[i0*8+n] + H[i1*8+n] + H[i2*8+n]) / 3
// Pure streaming: one float4 per lane, 8 rows per 256-thread block.
// ---------------------------------------------------------------------------
__global__ __launch_bounds__(256) void gather_kernel(
    const float* __restrict__ up,   // (UROWS, 128)
    const int*   __restrict__ idx,  // (UROWS, 3)
    const float* __restrict__ H,    // (MROWS, 128)
    float* __restrict__ out)        // (UROWS, 128)
{
    const int row = blockIdx.x * 8 + (threadIdx.x >> 5);   // 0..UROWS-1
    const int c   = (threadIdx.x & 31) * 4;                // float offset in row
    const int n   = row & (NN - 1);                        // row = lu*8 + n

    const int* __restrict__ ip = idx + (size_t)row * KNN;
    const int i0 = ip[0], i1 = ip[1], i2 = ip[2];

    const size_t h0 = ((size_t)i0 * NN + n) * D_OUT + c;
    const size_t h1r = ((size_t)i1 * NN + n) * D_OUT + c;
    const size_t h2 = ((size_t)i2 * NN + n) * D_OUT + c;

    v4f g0 = *(const v4f*)(H + h0);
    v4f g1 = *(const v4f*)(H + h1r);
    v4f g2 = *(const v4f*)(H + h2);
    v4f u  = *(const v4f*)(up + (size_t)row * D_OUT + c);

    const float inv3 = 1.0f / 3.0f;
    v4f r;
    #pragma unroll
    for (int j = 0; j < 4; ++j) r[j] = u[j] + (g0[j] + g1[j] + g2[j]) * inv3;

    *(v4f*)(out + (size_t)row * D_OUT + c) = r;
}

// ---------------------------------------------------------------------------
extern "C" void kernel_launch(void* const* d_in, const int* in_sizes, int n_in,
                              void* d_out, int out_size, void* d_ws, size_t ws_size,
                              hipStream_t stream) {
    const float* down = (const float*)d_in[0];
    const float* up   = (const float*)d_in[1];
    const int*   idx  = (const int*)d_in[2];
    const float* W1   = (const float*)d_in[3];
    const float* b1   = (const float*)d_in[4];
    const float* W2   = (const float*)d_in[5];
    const float* b2   = (const float*)d_in[6];
    float* out = (float*)d_out;
    float* H   = (float*)d_ws;   // needs MROWS*128*4 = 64 MB scratch

    mlp2_kernel<<<MROWS / 16, 256, 0, stream>>>(down, W1, b1, W2, b2, H);
    gather_kernel<<<UROWS / 8, 256, 0, stream>>>(up, idx, H, out);
}